// Quanvolution__gen168_65481071397074
// MI455X (gfx1250) — compile-verified
//
#include <hip/hip_runtime.h>
#include <hip/hip_bf16.h>

typedef __attribute__((ext_vector_type(2))) float v2f;
typedef __attribute__((ext_vector_type(8))) float v8f;

// ---- quantum pair: 2 layers of (RY x RY) then CNOT, real amplitudes ----
__device__ __forceinline__ void pair_expvals(const float* __restrict__ params,
                                             int w0, int w1,
                                             float& z0, float& z1) {
    float st0 = 1.f, st1 = 0.f, st2 = 0.f, st3 = 0.f;
#pragma unroll
    for (int layer = 0; layer < 2; ++layer) {
        float t0 = params[layer * 4 + w0] * 0.5f;
        float t1 = params[layer * 4 + w1] * 0.5f;
        float c0 = cosf(t0), s0 = sinf(t0);
        float c1 = cosf(t1), s1 = sinf(t1);
        // U = kron([[c0,-s0],[s0,c0]], [[c1,-s1],[s1,c1]])
        float n0 = c0*c1*st0 - c0*s1*st1 - s0*c1*st2 + s0*s1*st3;
        float n1 = c0*s1*st0 + c0*c1*st1 - s0*s1*st2 - s0*c1*st3;
        float n2 = s0*c1*st0 - s0*s1*st1 + c0*c1*st2 - c0*s1*st3;
        float n3 = s0*s1*st0 + s0*c1*st1 + c0*s1*st2 + c0*c1*st3;
        // CNOT(w0->w1): swap amplitudes 2 and 3
        st0 = n0; st1 = n1; st2 = n3; st3 = n2;
    }
    float p0 = st0*st0, p1 = st1*st1, p2 = st2*st2, p3 = st3*st3;
    z0 = p0 + p1 - p2 - p3;
    z1 = p0 - p1 + p2 - p3;
}

// ---- kernel 1: one wave32. Fold W over 196 patch repeats, contract with ev
// via V_WMMA_F32_16X16X4_F32, then log-softmax of the 10 logits -> res[10].
__global__ __launch_bounds__(32)
void quanv_logits_wmma_kernel(const float* __restrict__ params,
                              const float* __restrict__ W,
                              const float* __restrict__ bias,
                              float* __restrict__ res) {
    const int lane = threadIdx.x;          // 0..31, wave32
    // ev[4]: expvals; pairs (0,1) and (2,3) are decoupled
    float e0, e1, e2, e3;
    pair_expvals(params, 0, 1, e0, e1);
    pair_expvals(params, 2, 3, e2, e3);

    // A-matrix 16x4 f32: lanes 0-15 -> M=lane, v0/v1 = K0/K1;
    //                    lanes 16-31 -> M=lane-16, v0/v1 = K2/K3.
    const int m     = lane & 15;
    const int kbase = (lane < 16) ? 0 : 2;
    float a0 = 0.f, a1 = 0.f;
    if (m < 10) {                           // rows 10..15 are zero padding
        const float* wr = W + m * 784 + kbase;
#pragma unroll 4
        for (int c = 0; c < 196; ++c) {     // Wsum[m][k] = sum_p W[m][4p+k]
            a0 += wr[c * 4 + 0];
            a1 += wr[c * 4 + 1];
        }
    }
    v2f A;  A.x = a0; A.y = a1;
    // B-matrix 4x16 f32, column-uniform: B[k][n] = ev[k] for all n.
    // Layout: lanes 0-15 hold K=0(v0)/K=1(v1); lanes 16-31 hold K=2(v0)/K=3(v1).
    v2f Bm; Bm.x = (lane < 16) ? e0 : e2;
            Bm.y = (lane < 16) ? e1 : e3;
    v8f C = {};
    // D[m][n] = sum_k Wsum[m][k] * ev[k]  (identical across all 16 columns)
    C = __builtin_amdgcn_wmma_f32_16x16x4_f32(
            /*neg_a=*/false, A, /*neg_b=*/false, Bm,
            /*c_mod=*/(short)0, C, /*reuse_a=*/false, /*reuse_b=*/false);

    // C/D layout: lane 0 (N=0) holds M=0..7 in C[0..7]; lane 16 holds M=8..15.
    __shared__ float sl[16];
    if (lane == 0) {
#pragma unroll
        for (int r = 0; r < 8; ++r) sl[r] = C[r];
    }
    if (lane == 16) {
#pragma unroll
        for (int r = 0; r < 8; ++r) sl[8 + r] = C[r];
    }
    __syncthreads();

    if (lane == 0) {
        float l[10];
        float mx = -3.402823466e38f;
#pragma unroll
        for (int j = 0; j < 10; ++j) {
            l[j] = sl[j] + bias[j];
            mx = fmaxf(mx, l[j]);
        }
        float s = 0.f;
#pragma unroll
        for (int j = 0; j < 10; ++j) s += expf(l[j] - mx);
        float lse = mx + logf(s);
#pragma unroll
        for (int j = 0; j < 10; ++j) res[j] = l[j] - lse;
    }
}

// ---- kernel 2: broadcast res[10] to out[B][10] with float4 stores.
// Pattern period lcm(10,4) = 20 floats = 5 float4s, staged in LDS.
__global__ __launch_bounds__(256)
void quanv_bcast_kernel(const float* __restrict__ res,
                        float4* __restrict__ out4,
                        int n4, int tail, float* __restrict__ out_scalar) {
    __shared__ float pat[20];
    if (threadIdx.x < 20) pat[threadIdx.x] = res[threadIdx.x % 10];
    __syncthreads();
    int idx = blockIdx.x * blockDim.x + threadIdx.x;
    if (idx < n4) {
        int ph = (idx % 5) * 4;
        out4[idx] = make_float4(pat[ph], pat[ph + 1], pat[ph + 2], pat[ph + 3]);
    }
    if (blockIdx.x == 0 && threadIdx.x == 0 && tail) {
        for (int t = 0; t < tail; ++t) {
            int e = n4 * 4 + t;
            out_scalar[e] = res[e % 10];
        }
    }
}

extern "C" void kernel_launch(void* const* d_in, const int* in_sizes, int n_in,
                              void* d_out, int out_size, void* d_ws, size_t ws_size,
                              hipStream_t stream) {
    // inputs (setup_inputs order): x [B,1,28,28] (dead), params [2,4],
    //                              W [10,784], b [10]
    const float* params = (const float*)d_in[1];
    const float* W      = (const float*)d_in[2];
    const float* bias   = (const float*)d_in[3];
    float* res = (float*)d_ws;              // 10 floats of scratch
    float* out = (float*)d_out;             // [B,10] f32

    quanv_logits_wmma_kernel<<<1, 32, 0, stream>>>(params, W, bias, res);

    int n4   = out_size / 4;
    int tail = out_size % 4;
    int blocks = (n4 + 255) / 256;
    if (blocks < 1) blocks = 1;
    quanv_bcast_kernel<<<blocks, 256, 0, stream>>>(res, (float4*)out, n4, tail, out);
}